// MAFrAT_38706245272381
// MI455X (gfx1250) — compile-verified
//
#include <hip/hip_runtime.h>
#include <math.h>

// ---------------------------------------------------------------------------
// MAFrAT forward for MI455X (gfx1250, wave32).
// FrFT chains = batched complex GEMMs on the FP32 WMMA pipe
// (v_wmma_f32_16x16x4_f32, f32 in / f32 acc -- right precision for an f32
// reference, and the op is HBM-bound so the FP32 matrix rate is free).
// A-tiles optionally staged to LDS by the Tensor Data Mover.
// ---------------------------------------------------------------------------

typedef __attribute__((ext_vector_type(8))) float v8f;
typedef __attribute__((ext_vector_type(4))) float v4f;
typedef __attribute__((ext_vector_type(2))) float v2f;
typedef __attribute__((ext_vector_type(4))) unsigned int v4u;
typedef __attribute__((ext_vector_type(8))) int v8i;
typedef __attribute__((ext_vector_type(4))) int v4i;

#define BB   8
#define DIMC 64
#define HW   160
#define D4   16
#define WPB  2   // waves (16-wide N tiles) per block -> N=160 tiles exactly

// ---------------- Tensor Data Mover: 2-D f32 tile -> LDS --------------------
#if __has_builtin(__builtin_amdgcn_tensor_load_to_lds)
#define HAVE_TDM 1
__device__ __forceinline__ void tdm_load_tile_2d(void* ldsDst, const void* gsrc,
                                                 int tensorD0, int tensorD1,
                                                 long strideD0) {
  unsigned long long ga = (unsigned long long)(uintptr_t)gsrc;
  unsigned int lds = (unsigned int)(uintptr_t)ldsDst;  // LDS aperture: low 32b
  v4u g0;
  g0.x = 1u;                                           // count=1 (user D#)
  g0.y = lds;                                          // lds_addr
  g0.z = (unsigned int)(ga & 0xffffffffull);           // global_addr[31:0]
  g0.w = (unsigned int)((ga >> 32) & 0x1ffffffull) | (2u << 30);  // [56:32]|type=2
  v8i g1;
  g1[0] = (2 << 16);                                   // data_size = 4 bytes
  g1[1] = (int)((tensorD0 & 0xffff) << 16);            // tensor_dim0 lo16
  g1[2] = (int)(((tensorD0 >> 16) & 0xffff) | ((tensorD1 & 0xffff) << 16));
  g1[3] = (int)(((tensorD1 >> 16) & 0xffff) | (16 << 16));  // tile_dim0 = 16
  g1[4] = 16;                                          // tile_dim1 = 16
  g1[5] = (int)(strideD0 & 0xffffffffll);              // dim0 stride lo32
  g1[6] = (int)((strideD0 >> 32) & 0xffffll);          // stride hi, dim1_stride=0
  g1[7] = 0;
  v4i z4 = {0, 0, 0, 0};
#if defined(__clang_major__) && __clang_major__ >= 23
  v8i z8 = {0, 0, 0, 0, 0, 0, 0, 0};
  __builtin_amdgcn_tensor_load_to_lds(g0, g1, z4, z4, z8, 0);
#else
  __builtin_amdgcn_tensor_load_to_lds(g0, g1, z4, z4, 0);
#endif
}
#else
#define HAVE_TDM 0
#endif

// ---------------- reduction for frac order p --------------------------------
__global__ void reduce_partial_kernel(const float* __restrict__ x, long n,
                                      float* __restrict__ part) {
  __shared__ float sm[256];
  float s = 0.f;
  for (long i = blockIdx.x * 256L + threadIdx.x; i < n; i += 256L * gridDim.x)
    s += x[i];
  sm[threadIdx.x] = s;
  __syncthreads();
  for (int o = 128; o > 0; o >>= 1) {
    if ((int)threadIdx.x < o) sm[threadIdx.x] += sm[threadIdx.x + o];
    __syncthreads();
  }
  if (threadIdx.x == 0) part[blockIdx.x] = sm[0];
}

__global__ void reduce_final_kernel(const float* __restrict__ part, long n,
                                    float* __restrict__ pfrac) {
  __shared__ float sm[256];
  sm[threadIdx.x] = part[threadIdx.x];
  __syncthreads();
  for (int o = 128; o > 0; o >>= 1) {
    if ((int)threadIdx.x < o) sm[threadIdx.x] += sm[threadIdx.x + o];
    __syncthreads();
  }
  if (threadIdx.x == 0) {
    float mean = sm[0] / (float)n;
    pfrac[0] = 0.5f + 0.4f * tanhf(mean);  // bounded fractional order
  }
}

// ---------------- DFrFT kernel matrix build (complex f32 planes) ------------
__global__ void build_frft_kernel(float* __restrict__ Kr, float* __restrict__ Ki,
                                  int N, const float* __restrict__ pfrac,
                                  float sign) {
  int idx = blockIdx.x * blockDim.x + threadIdx.x;
  if (idx >= N * N) return;
  int m = idx / N, n = idx % N;
  float p = sign * pfrac[0];
  float alpha = p * 1.57079632679f;
  float s = sinf(alpha), c = cosf(alpha);
  if (fabsf(s) < 1e-6f) s = (s >= 0.f ? 1e-6f : -1e-6f);
  float cot = c / s, csc = 1.0f / s;
  float zr = 1.0f / (float)N, zi = -cot / (float)N;   // (1 - i cot)/N
  float r = sqrtf(zr * zr + zi * zi);
  float th = 0.5f * atan2f(zi, zr);
  float sr = sqrtf(r);
  float ar = sr * cosf(th), ai = sr * sinf(th);       // complex sqrt
  float fm = (float)m, fn = (float)n;
  float ph = (3.14159265359f / (float)N) *
             (cot * (fm * fm + fn * fn) - 2.0f * csc * fm * fn);
  float cr = cosf(ph), ci = sinf(ph);
  Kr[idx] = ar * cr - ai * ci;
  Ki[idx] = ar * ci + ai * cr;
}

// ---------------- batched complex GEMM, FP32 WMMA pipe ----------------------
// C[bat](MxN) = A[bat](MxK) x B[bat](KxN | transB: stored NxK row-major).
// Requirements used here: M,N,K multiples of 16; N tiles exactly (WPB waves).
__global__ __launch_bounds__(64) void cgemm_f32_wmma_kernel(
    const float* __restrict__ Ar, const float* __restrict__ Ai,
    const float* __restrict__ Br, const float* __restrict__ Bi,
    float* __restrict__ Cr, float* __restrict__ Ci,
    int M, int N, int K, long sA, long sB, long sC, int transB, int useTdm) {
  __shared__ float sAr_[16 * 16], sAi_[16 * 16];
  __shared__ float sBr_[WPB][16 * 16], sBi_[WPB][16 * 16];

  const int tid = threadIdx.x;
  const int lane = tid & 31, wv = tid >> 5;
  const int hi = lane >> 4, lm = lane & 15;
  const int bat = blockIdx.z;
  const int m0 = blockIdx.x * 16;
  const int n0 = (blockIdx.y * WPB + wv) * 16;
  (void)useTdm;

  const float* Arb = Ar + (long)bat * sA;
  const float* Aib = Ai + (long)bat * sA;
  const float* Brb = Br + (long)bat * sB;
  const float* Bib = Bi + (long)bat * sB;

  v8f accRR = {}, accII = {}, accRI = {}, accIR = {};

  for (int k0 = 0; k0 < K; k0 += 16) {
    if (k0 + 16 < K)  // hint next B slab toward caches (global_prefetch_b8)
      __builtin_prefetch(Brb + (transB ? (long)n0 * K + k0 + 16
                                       : (long)(k0 + 16) * N + n0), 0, 1);
#if HAVE_TDM
    if (useTdm) {
      if (wv == 0) {  // one wave drives the Tensor Data Mover for the A tile
        tdm_load_tile_2d(sAr_, Arb + (long)m0 * K + k0, K, M, (long)K);
        tdm_load_tile_2d(sAi_, Aib + (long)m0 * K + k0, K, M, (long)K);
        __builtin_amdgcn_s_wait_tensorcnt(0);
      }
    } else
#endif
    {
      // A slab 16x16 f32 per plane: 128 float4 chunks over 64 threads
#pragma unroll
      for (int j = 0; j < 2; ++j) {
        int ch = tid + j * 64;               // 0..127
        int pl = ch >> 6, idx = ch & 63;
        int am = idx >> 2, cq = idx & 3;
        const float* src = (pl ? Aib : Arb) + (long)(m0 + am) * K + k0 + cq * 4;
        v4f v = *(const v4f*)src;
        float* dst = (pl ? sAi_ : sAr_) + am * 16 + cq * 4;
        *(v4f*)dst = v;
      }
    }
    // B slab 16x16 f32 per plane per wave: 128 chunks over 32 lanes
    if (!transB) {
#pragma unroll
      for (int j = 0; j < 4; ++j) {
        int ch = lane + j * 32;
        int pl = ch >> 6, idx = ch & 63;
        int bk = idx >> 2, cq = idx & 3;
        const float* src = (pl ? Bib : Brb) + (long)(k0 + bk) * N + n0 + cq * 4;
        v4f v = *(const v4f*)src;
        float* dst = (pl ? sBi_[wv] : sBr_[wv]) + bk * 16 + cq * 4;
        *(v4f*)dst = v;
      }
    } else {  // plane NxK: contiguous along k -> vector load, scatter to [k][n]
#pragma unroll
      for (int j = 0; j < 4; ++j) {
        int ch = lane + j * 32;
        int pl = ch >> 6, idx = ch & 63;
        int bn = idx >> 2, cq = idx & 3;
        const float* src = (pl ? Bib : Brb) + (long)(n0 + bn) * K + k0 + cq * 4;
        v4f v = *(const v4f*)src;
        float* dstp = (pl ? sBi_[wv] : sBr_[wv]);
        dstp[(cq * 4 + 0) * 16 + bn] = v.x;
        dstp[(cq * 4 + 1) * 16 + bn] = v.y;
        dstp[(cq * 4 + 2) * 16 + bn] = v.z;
        dstp[(cq * 4 + 3) * 16 + bn] = v.w;
      }
    }
    __syncthreads();

    // 4 sub-steps of K=4, each 4 complex WMMA ops -> 16 v_wmma per slab
#pragma unroll
    for (int kk = 0; kk < 16; kk += 4) {
      int ka = kk + hi * 2;  // 16x4 f32 A layout: v0=K{0|2}, v1=K{1|3}
      v2f arf = *(const v2f*)&sAr_[lm * 16 + ka];
      v2f aif = *(const v2f*)&sAi_[lm * 16 + ka];
      v2f brf, bif;
      brf.x = sBr_[wv][(ka + 0) * 16 + lm];
      brf.y = sBr_[wv][(ka + 1) * 16 + lm];
      bif.x = sBi_[wv][(ka + 0) * 16 + lm];
      bif.y = sBi_[wv][(ka + 1) * 16 + lm];
      accRR = __builtin_amdgcn_wmma_f32_16x16x4_f32(false, arf, false, brf,
                                                    (short)0, accRR, false, false);
      accII = __builtin_amdgcn_wmma_f32_16x16x4_f32(false, aif, false, bif,
                                                    (short)0, accII, false, false);
      accRI = __builtin_amdgcn_wmma_f32_16x16x4_f32(false, arf, false, bif,
                                                    (short)0, accRI, false, false);
      accIR = __builtin_amdgcn_wmma_f32_16x16x4_f32(false, aif, false, brf,
                                                    (short)0, accIR, false, false);
    }
    __syncthreads();
  }

  float* Crb = Cr + (long)bat * sC;
  float* Cib = Ci + (long)bat * sC;
#pragma unroll
  for (int r = 0; r < 8; ++r) {
    int m = r + hi * 8;  // f32 C/D layout: VGPR r -> row r (+8 for hi half)
    long idx = (long)(m0 + m) * N + (n0 + lm);
    Crb[idx] = accRR[r] - accII[r];
    Cib[idx] = accRI[r] + accIR[r];
  }
}

// ---------------- gather x -> planar complex f32 ----------------------------
__global__ void gather_branch_kernel(const float* __restrict__ x,
                                     float* __restrict__ Ur,
                                     float* __restrict__ Ui, int mode) {
  long t = blockIdx.x * (long)blockDim.x + threadIdx.x;
  const long total = (long)BB * D4 * HW * HW;
  if (t >= total) return;
  int w = (int)(t % HW);
  long r1 = t / HW;
  int h = (int)(r1 % HW);
  long r2 = r1 / HW;
  int c = (int)(r2 % D4);
  int b = (int)(r2 / D4);
  long src, dst;
  if (mode == 0) {  // y1: xb(b,h,c,w), channels [0,16)
    src = (((long)(b * DIMC + c) * HW + h) * HW + w);
    dst = ((long)(b * HW + h) * (D4 * HW) + c * HW + w);
  } else if (mode == 1) {  // y2: xb(b,w,c,h), channels [16,32)
    src = (((long)(b * DIMC + 16 + c) * HW + h) * HW + w);
    dst = ((long)(b * HW + w) * (D4 * HW) + c * HW + h);
  } else {  // y3: (b,c,h,w), channels [32,48)
    src = (((long)(b * DIMC + 32 + c) * HW + h) * HW + w);
    dst = ((long)(b * D4 + c) * (HW * HW) + (long)h * HW + w);
  }
  Ur[dst] = x[src];
  Ui[dst] = 0.f;
}

// ---------------- complex spectral weight multiply --------------------------
__global__ void mul_weight_1d_kernel(float* __restrict__ Ur,
                                     float* __restrict__ Ui,
                                     const float* __restrict__ tab) {
  long t = blockIdx.x * (long)blockDim.x + threadIdx.x;
  const long total = (long)BB * HW * D4 * HW;  // bat(1280) x 16 x 160
  if (t >= total) return;
  int n = (int)(t % HW);
  long r = t / HW;
  int m = (int)(r % D4);
  float wr = 1.f, wi = 0.f;
  if (tab) {
    float xs = n * (15.0f / 159.0f);  // align-corners 16 -> 160
    int x0 = (int)floorf(xs);
    float fx = xs - (float)x0;
    int x1 = (x0 + 1 < 15) ? x0 + 1 : 15;
    wr = tab[m * 16 + x0] * (1.f - fx) + tab[m * 16 + x1] * fx;
    wi = tab[256 + m * 16 + x0] * (1.f - fx) + tab[256 + m * 16 + x1] * fx;
  }
  float ur = Ur[t], ui = Ui[t];
  Ur[t] = ur * wr - ui * wi;
  Ui[t] = ur * wi + ui * wr;
}

__global__ void mul_weight_2d_kernel(float* __restrict__ Ur,
                                     float* __restrict__ Ui,
                                     const float* __restrict__ tab) {
  long t = blockIdx.x * (long)blockDim.x + threadIdx.x;
  const long total = (long)BB * D4 * HW * HW;  // bat(128) x 160 x 160
  if (t >= total) return;
  int w = (int)(t % HW);
  long r = t / HW;
  int h = (int)(r % HW);
  long bat = r / HW;
  int c = (int)(bat % D4);
  float wr = 1.f, wi = 0.f;
  if (tab) {
    float ys = h * (15.0f / 159.0f), xs = w * (15.0f / 159.0f);
    int y0 = (int)floorf(ys), x0 = (int)floorf(xs);
    float fy = ys - (float)y0, fx = xs - (float)x0;
    int y1 = (y0 + 1 < 15) ? y0 + 1 : 15;
    int x1 = (x0 + 1 < 15) ? x0 + 1 : 15;
    const float* tr = tab + c * 256;
    const float* ti = tab + (16 + c) * 256;
    wr = tr[y0 * 16 + x0] * (1 - fy) * (1 - fx) + tr[y0 * 16 + x1] * (1 - fy) * fx +
         tr[y1 * 16 + x0] * fy * (1 - fx) + tr[y1 * 16 + x1] * fy * fx;
    wi = ti[y0 * 16 + x0] * (1 - fy) * (1 - fx) + ti[y0 * 16 + x1] * (1 - fy) * fx +
         ti[y1 * 16 + x0] * fy * (1 - fx) + ti[y1 * 16 + x1] * fy * fx;
  }
  float ur = Ur[t], ui = Ui[t];
  Ur[t] = ur * wr - ui * wi;
  Ui[t] = ur * wi + ui * wr;
}

// ---------------- |z| + scatter into output layout --------------------------
__global__ void abs_scatter_kernel(const float* __restrict__ Ur,
                                   const float* __restrict__ Ui,
                                   float* __restrict__ out, int mode) {
  long t = blockIdx.x * (long)blockDim.x + threadIdx.x;
  const long total = (long)BB * D4 * HW * HW;
  if (t >= total) return;
  float cr = Ur[t], ci = Ui[t];
  float mag = sqrtf(cr * cr + ci * ci);
  long dst;
  if (mode == 0) {  // U(b,h,c,w) -> out[b, c, h, w]
    int w = (int)(t % HW);
    long r = t / HW;
    int c = (int)(r % D4);
    long r2 = r / D4;
    int h = (int)(r2 % HW);
    int b = (int)(r2 / HW);
    dst = (((long)(b * DIMC + c) * HW + h) * HW + w);
  } else if (mode == 1) {  // U(b,w,c,h) -> out[b, 16+c, w, h]
    int h = (int)(t % HW);
    long r = t / HW;
    int c = (int)(r % D4);
    long r2 = r / D4;
    int w = (int)(r2 % HW);
    int b = (int)(r2 / HW);
    dst = (((long)(b * DIMC + 16 + c) * HW + w) * HW + h);
  } else {  // U(b,c,h,w) -> out[b, 32+c, h, w]
    int w = (int)(t % HW);
    long r = t / HW;
    int h = (int)(r % HW);
    long r2 = r / HW;
    int c = (int)(r2 % D4);
    int b = (int)(r2 / D4);
    dst = (((long)(b * DIMC + 32 + c) * HW + h) * HW + w);
  }
  out[dst] = mag;
}

// ---------------- y4 branch (idw residual, elementwise form) ----------------
__global__ void y4_kernel(const float* __restrict__ x, float* __restrict__ out) {
  long t = blockIdx.x * (long)blockDim.x + threadIdx.x;
  const long total = (long)BB * D4 * HW * HW;
  if (t >= total) return;
  int w = (int)(t % HW);
  long r = t / HW;
  int h = (int)(r % HW);
  long r2 = r / HW;
  int c = (int)(r2 % D4);
  int b = (int)(r2 / D4);
  long idx = (((long)(b * DIMC + 48 + c) * HW + h) * HW + w);
  float v = x[idx];
  float g = 0.5f * v * (1.0f + erff(v * 0.70710678f));
  out[idx] = v + 0.1f * g;
}

// ---------------------------------------------------------------------------
extern "C" void kernel_launch(void* const* d_in, const int* in_sizes, int n_in,
                              void* d_out, int out_size, void* d_ws,
                              size_t ws_size, hipStream_t stream) {
  const float* x = (const float*)d_in[0];
  float* out = (float*)d_out;
  const long nX = (long)BB * DIMC * HW * HW;

  // locate spectral weight tables by element count (a/b: 512, c: 8192)
  const float* aW = nullptr;
  const float* bW = nullptr;
  const float* cW = nullptr;
  for (int i = 1; i < n_in; ++i) {
    if (in_sizes[i] == 512) {
      if (!aW) aW = (const float*)d_in[i];
      else if (!bW) bW = (const float*)d_in[i];
    } else if (in_sizes[i] == 8192 && !cW) {
      cW = (const float*)d_in[i];
    }
  }

  size_t off = 0;
  char* wsb = (char*)d_ws;
  auto alloc = [&](size_t bytes) -> void* {
    void* p = wsb + off;
    off = (off + bytes + 255) & ~(size_t)255;
    return p;
  };
  float* part = (float*)alloc(256 * sizeof(float));
  float* pfrac = (float*)alloc(sizeof(float));
  const size_t k160 = (size_t)HW * HW;
  const size_t k16 = (size_t)D4 * D4;
  float* K160p_r = (float*)alloc(k160 * 4);
  float* K160p_i = (float*)alloc(k160 * 4);
  float* K160m_r = (float*)alloc(k160 * 4);
  float* K160m_i = (float*)alloc(k160 * 4);
  float* K16p_r = (float*)alloc(k16 * 4);
  float* K16p_i = (float*)alloc(k16 * 4);
  float* K16m_r = (float*)alloc(k16 * 4);
  float* K16m_i = (float*)alloc(k16 * 4);
  const size_t plane = (size_t)BB * D4 * HW * HW;  // 3,276,800
  float* Ur = (float*)alloc(plane * 4);
  float* Ui = (float*)alloc(plane * 4);
  float* Vr = (float*)alloc(plane * 4);
  float* Vi = (float*)alloc(plane * 4);
  if (off > ws_size) return;

  reduce_partial_kernel<<<256, 256, 0, stream>>>(x, nX, part);
  reduce_final_kernel<<<1, 256, 0, stream>>>(part, nX, pfrac);

  build_frft_kernel<<<(int)((k160 + 255) / 256), 256, 0, stream>>>(
      K160p_r, K160p_i, HW, pfrac, +1.f);
  build_frft_kernel<<<(int)((k160 + 255) / 256), 256, 0, stream>>>(
      K160m_r, K160m_i, HW, pfrac, -1.f);
  build_frft_kernel<<<1, 256, 0, stream>>>(K16p_r, K16p_i, D4, pfrac, +1.f);
  build_frft_kernel<<<1, 256, 0, stream>>>(K16m_r, K16m_i, D4, pfrac, -1.f);

  const int eltBlocks = (int)((plane + 255) / 256);
  dim3 blk(64, 1, 1);
  auto gemm = [&](const float* arp, const float* aip, const float* brp,
                  const float* bip, float* crp, float* cip, int M, int N, int K,
                  long sA, long sB, long sC, int tB, int batch, int tdm) {
    dim3 grd(M / 16, N / 16 / WPB, batch);
    cgemm_f32_wmma_kernel<<<grd, blk, 0, stream>>>(arp, aip, brp, bip, crp, cip,
                                                   M, N, K, sA, sB, sC, tB, tdm);
  };

  // ---------- y1: K16.X.K160^T chain, batch = B*HW = 1280 -------------------
  gather_branch_kernel<<<eltBlocks, 256, 0, stream>>>(x, Ur, Ui, 0);
  gemm(K16p_r, K16p_i, Ur, Ui, Vr, Vi, D4, HW, D4, 0, (long)D4 * HW,
       (long)D4 * HW, 0, BB * HW, 0);
  gemm(Vr, Vi, K160p_r, K160p_i, Ur, Ui, D4, HW, HW, (long)D4 * HW, 0,
       (long)D4 * HW, 1, BB * HW, 0);
  mul_weight_1d_kernel<<<eltBlocks, 256, 0, stream>>>(Ur, Ui, aW);
  gemm(K16m_r, K16m_i, Ur, Ui, Vr, Vi, D4, HW, D4, 0, (long)D4 * HW,
       (long)D4 * HW, 0, BB * HW, 0);
  gemm(Vr, Vi, K160m_r, K160m_i, Ur, Ui, D4, HW, HW, (long)D4 * HW, 0,
       (long)D4 * HW, 1, BB * HW, 0);
  abs_scatter_kernel<<<eltBlocks, 256, 0, stream>>>(Ur, Ui, out, 0);

  // ---------- y2: same chain on (b,w,c,h) view ------------------------------
  gather_branch_kernel<<<eltBlocks, 256, 0, stream>>>(x, Ur, Ui, 1);
  gemm(K16p_r, K16p_i, Ur, Ui, Vr, Vi, D4, HW, D4, 0, (long)D4 * HW,
       (long)D4 * HW, 0, BB * HW, 0);
  gemm(Vr, Vi, K160p_r, K160p_i, Ur, Ui, D4, HW, HW, (long)D4 * HW, 0,
       (long)D4 * HW, 1, BB * HW, 0);
  mul_weight_1d_kernel<<<eltBlocks, 256, 0, stream>>>(Ur, Ui, bW);
  gemm(K16m_r, K16m_i, Ur, Ui, Vr, Vi, D4, HW, D4, 0, (long)D4 * HW,
       (long)D4 * HW, 0, BB * HW, 0);
  gemm(Vr, Vi, K160m_r, K160m_i, Ur, Ui, D4, HW, HW, (long)D4 * HW, 0,
       (long)D4 * HW, 1, BB * HW, 0);
  abs_scatter_kernel<<<eltBlocks, 256, 0, stream>>>(Ur, Ui, out, 1);

  // ---------- y3: 2-D FrFT, K160.X.K160^T, batch = B*D4 = 128 (TDM A-tiles) -
  gather_branch_kernel<<<eltBlocks, 256, 0, stream>>>(x, Ur, Ui, 2);
  gemm(K160p_r, K160p_i, Ur, Ui, Vr, Vi, HW, HW, HW, 0, (long)HW * HW,
       (long)HW * HW, 0, BB * D4, 1);
  gemm(Vr, Vi, K160p_r, K160p_i, Ur, Ui, HW, HW, HW, (long)HW * HW, 0,
       (long)HW * HW, 1, BB * D4, 1);
  mul_weight_2d_kernel<<<eltBlocks, 256, 0, stream>>>(Ur, Ui, cW);
  gemm(K160m_r, K160m_i, Ur, Ui, Vr, Vi, HW, HW, HW, 0, (long)HW * HW,
       (long)HW * HW, 0, BB * D4, 1);
  gemm(Vr, Vi, K160m_r, K160m_i, Ur, Ui, HW, HW, HW, (long)HW * HW, 0,
       (long)HW * HW, 1, BB * D4, 1);
  abs_scatter_kernel<<<eltBlocks, 256, 0, stream>>>(Ur, Ui, out, 2);

  // ---------- y4 ------------------------------------------------------------
  y4_kernel<<<eltBlocks, 256, 0, stream>>>(x, out);
}